// LIF_27590869910127
// MI455X (gfx1250) — compile-verified
//
#include <hip/hip_runtime.h>
#include <stdint.h>

// ---------------- problem constants (reference: B=2048, T=4096) -------------
#define ROWS   32                 // rows per block == lanes per wave32
#define T_LEN  4096
#define TT     64                 // timesteps per tile (must be power-of-2 for TDM pad interval)
#define PAD    4                  // 4 dwords (16B) LDS row padding -> stride 68 floats
#define RSTR   (TT + PAD)
#define NT     (T_LEN / TT)       // 64 tiles
#define CH     (TT / 4)           // 16 float4 chunks per tile
#define NBUF   4                  // input tile pipeline depth (TENSORcnt)
#define VTHR   1.27f
#define MBIG   (1 << 20)

typedef uint32_t u32;
typedef uint64_t u64;
typedef __attribute__((ext_vector_type(4))) u32 u32x4_t;
typedef __attribute__((ext_vector_type(8))) int i32x8_t;
typedef __attribute__((ext_vector_type(4))) int i32x4_t;

#if defined(__HIP_DEVICE_COMPILE__) && __has_builtin(__builtin_amdgcn_tensor_load_to_lds) && __has_builtin(__builtin_amdgcn_s_wait_tensorcnt)
#define HAVE_TDM 1
#else
#define HAVE_TDM 0
#endif

#if HAVE_TDM
__device__ __forceinline__ u32 ldsOffset(const void* p) {
  // generic LDS address low 32 bits == LDS byte offset (CDNA5 flat aperture rule)
  return (u32)(uintptr_t)p;
}

// One TDM op: 2D tile [32 rows x TT cols], global row stride T_LEN elements,
// linear into LDS with 16B padding after every TT dwords (row stride RSTR floats).
__device__ __forceinline__ void tdm_load(u32 lds_addr, u64 gaddr) {
  u32x4_t g0;
  g0[0] = 1u;                                   // count=1 valid descriptor
  g0[1] = lds_addr;                             // lds_addr (bytes)
  g0[2] = (u32)gaddr;                           // global_addr[31:0]
  g0[3] = (u32)((gaddr >> 32) & 0x1FFFFFFull)   // global_addr[56:32]
        | (2u << 30);                           // type = 2 ("image")
  i32x8_t g1;
  g1[0] = (2 << 16)     // data_size = 4B
        | (1 << 20)     // pad_enable
        | (5 << 22)     // pad_interval code 5 -> 64 dwords (== TT)
        | (3 << 25);    // pad_amount   code 3 -> 4 dwords (16B)
  g1[1] = (TT   & 0xFFFF) << 16;  // tensor_dim0 lo16
  g1[2] = (ROWS & 0xFFFF) << 16;  // tensor_dim0 hi16 (0) | tensor_dim1 lo16
  g1[3] = (TT   & 0xFFFF) << 16;  // tensor_dim1 hi16 (0) | tile_dim0
  g1[4] = ROWS;                   // tile_dim1 = 32, tile_dim2 = 0
  g1[5] = T_LEN;                  // tensor_dim0_stride lo32 (elements)
  g1[6] = 0;                      // stride hi16 | tensor_dim1_stride lo16
  g1[7] = 0;
  i32x4_t gz4 = {0, 0, 0, 0};                    // groups 2/3 unused (2D tile)
  i32x8_t gz8 = {0, 0, 0, 0, 0, 0, 0, 0};        // trailing group (clang-23 6-arg form)
  __builtin_amdgcn_tensor_load_to_lds(g0, g1, gz4, gz4, gz8, 0);
}
#define WAIT_TENSOR(n) __builtin_amdgcn_s_wait_tensorcnt((unsigned short)(n))
#endif

// ---------------- LIF step, reduced state {v, m, px} (exact for t>=4) -------
// m = steps since last masked (t>=1) spike; spike at this step -> m=0.
struct St { float v; int m; float px; };

__device__ __forceinline__ void lif_step(float x, St& st, float decay,
                                         float& vo, float& so, float& dvo) {
  float vn = __fsub_rn(st.v, __fmul_rn(decay, st.v)); // v - decay*(v-0), two roundings like ref
  float xe = (st.m <= 4) ? 0.0f : x;                  // z>0 window == m in [0,4]
  vn = __fadd_rn(vn, xe);
  vn = (st.m == 0) ? 0.0f : vn;                       // counter==1 reset (input already zeroed)
  bool sp = vn > VTHR;                                // spike(v - V_TH)
  vn = sp ? VTHR : vn;                                // counter>1 -> V_TH (only at spike step)
  st.v = vn;
  vo = vn;
  so = sp ? 1.0f : 0.0f;
  st.m = sp ? 0 : (st.m + 1);
  dvo = (st.m <= 6) ? 0.0f : st.px;                   // dv[t-1] zero iff spike in [t-6, t]
  st.px = x;
}

// Chunk of 4 steps at global column j (j >= 4). Emits dv group [j-4, j-1].
__device__ __forceinline__ void do_chunk(float4 x, St& st,
                                         float& dA, float& dB, float& dC,
                                         float* __restrict__ vp,
                                         float* __restrict__ sp_,
                                         float* __restrict__ dvp,
                                         int j, float decay) {
  float v0, v1, v2, v3, s0, s1, s2, s3, dD, nA, nB, nC;
  lif_step(x.x, st, decay, v0, s0, dD);
  float4 d4; d4.x = dA; d4.y = dB; d4.z = dC; d4.w = dD;
  *(float4*)(dvp + j - 4) = d4;
  lif_step(x.y, st, decay, v1, s1, nA);
  lif_step(x.z, st, decay, v2, s2, nB);
  lif_step(x.w, st, decay, v3, s3, nC);
  float4 v4; v4.x = v0; v4.y = v1; v4.z = v2; v4.w = v3;
  *(float4*)(vp + j) = v4;
  float4 s4; s4.x = s0; s4.y = s1; s4.z = s2; s4.w = s3;
  *(float4*)(sp_ + j) = s4;
  dA = nA; dB = nB; dC = nC;
}

// Exact reference semantics for t=0..3 (full counter/z state; absorbs the
// t=0/t=1 double-spike corner where counter can reach 2). Emits dv[0..2] into
// the lag registers, v/s for columns 0..3, and the reduced state for t>=4.
__device__ __forceinline__ void chunk0_special(float4 x, St& st,
                                               float& dA, float& dB, float& dC,
                                               float* __restrict__ vp,
                                               float* __restrict__ sp_,
                                               float decay) {
  // t = 0 : v=0 -> decay term 0; c==1 false; z==0; spike at t=0 not masked
  float v  = x.x;
  bool sp  = v > VTHR;
  float v0 = sp ? VTHR : v;
  float s0 = sp ? 1.0f : 0.0f;
  int   c  = sp ? 1 : 0;
  int   z  = 0;
  int   m  = MBIG;
  float px = x.x;
  v = v0;
  // t = 1 : z==0 so input active; c>1 impossible pre-spike
  float vn = __fsub_rn(v, __fmul_rn(decay, v));
  vn = (c == 1) ? 0.0f : vn;
  vn = __fadd_rn(vn, x.y);
  sp = vn > VTHR;
  float s1 = sp ? 1.0f : 0.0f;
  c += sp ? 2 : 0;
  vn = (c > 1) ? VTHR : vn;
  z  = sp ? 5 : 0;
  c  = (c - 1 > 0) ? (c - 1) : 0;
  m  = sp ? 0 : (m + 1);
  float d0 = (m <= 6) ? 0.0f : px;
  px = x.y; float v1 = vn; v = vn;
  // t = 2 : full semantics (pre-spike c>1 force can fire here)
  vn = __fsub_rn(v, __fmul_rn(decay, v));
  vn = (c == 1) ? 0.0f : vn;
  float xe = (z > 0) ? 0.0f : x.z;
  vn = __fadd_rn(vn, xe);
  vn = (c > 1) ? VTHR : vn;
  sp = vn > VTHR;
  float s2 = sp ? 1.0f : 0.0f;
  c += sp ? 2 : 0;
  vn = (c > 1) ? VTHR : vn;
  z  = sp ? 5 : ((z - 1 > 0) ? (z - 1) : 0);
  c  = (c - 1 > 0) ? (c - 1) : 0;
  m  = sp ? 0 : (m + 1);
  float d1 = (m <= 6) ? 0.0f : px;
  px = x.z; float v2 = vn; v = vn;
  // t = 3 : full semantics
  vn = __fsub_rn(v, __fmul_rn(decay, v));
  vn = (c == 1) ? 0.0f : vn;
  xe = (z > 0) ? 0.0f : x.w;
  vn = __fadd_rn(vn, xe);
  vn = (c > 1) ? VTHR : vn;
  sp = vn > VTHR;
  float s3 = sp ? 1.0f : 0.0f;
  c += sp ? 2 : 0;
  vn = (c > 1) ? VTHR : vn;
  m  = sp ? 0 : (m + 1);
  float d2 = (m <= 6) ? 0.0f : px;
  px = x.w; float v3 = vn; v = vn;

  st.v = v; st.m = m; st.px = px;
  dA = d0; dB = d1; dC = d2;
  float4 v4; v4.x = v0; v4.y = v1; v4.z = v2; v4.w = v3;
  *(float4*)(vp + 0) = v4;
  float4 s4; s4.x = s0; s4.y = s1; s4.z = s2; s4.w = s3;
  *(float4*)(sp_ + 0) = s4;
}

// ---------------------------------------------------------------------------
__global__ __launch_bounds__(ROWS, 1)
void lif_scan_kernel(const float* __restrict__ ode,
                     const float* __restrict__ decayPtr,
                     float* __restrict__ out, int B) {
#if HAVE_TDM
  __shared__ float sIn[NBUF][ROWS * RSTR];   // ~34.8 KB
#endif
  const int lane = threadIdx.x;
  const int row  = blockIdx.x * ROWS + lane;
  const float decay = decayPtr[0];

  const float* __restrict__ xrow = ode + (size_t)row * T_LEN;
  float* __restrict__ vout  = out + (size_t)row * T_LEN;
  float* __restrict__ souto = out + (size_t)B * T_LEN + (size_t)row * T_LEN;
  float* __restrict__ dvout = out + 2 * (size_t)B * T_LEN + (size_t)row * T_LEN;

#if HAVE_TDM
  const u64 gbase = (u64)(uintptr_t)(ode + (size_t)(blockIdx.x * ROWS) * T_LEN);
  #pragma unroll
  for (int k = 0; k < NBUF; ++k)
    tdm_load(ldsOffset(&sIn[k][0]), gbase + (u64)k * TT * 4u);
  #define XCHUNK(k, c) (((const float4*)&sIn[(k) & (NBUF - 1)][lane * RSTR])[(c)])
#else
  #define XCHUNK(k, c) (*(const float4*)(xrow + (size_t)(k) * TT + 4 * (c)))
#endif
  (void)xrow;

  St st; float dA, dB, dC;

  // ----- tile 0 (prologue steps t=0..3 use exact reference semantics) -----
  {
#if HAVE_TDM
    WAIT_TENSOR(NBUF - 1);        // loads in order -> tile 0 resident
#endif
    float4 x0 = XCHUNK(0, 0);
    chunk0_special(x0, st, dA, dB, dC, vout, souto, decay);
    #pragma unroll
    for (int c = 1; c < CH; ++c) {
      float4 xx = XCHUNK(0, c);
      do_chunk(xx, st, dA, dB, dC, vout, souto, dvout, 4 * c, decay);
    }
#if HAVE_TDM
    if (NBUF < NT)
      tdm_load(ldsOffset(&sIn[0][0]), gbase + (u64)NBUF * TT * 4u);
#endif
  }

  // ----- tiles 1 .. NT-1 -----
  for (int k = 1; k < NT; ++k) {
#if HAVE_TDM
    WAIT_TENSOR(NBUF - 1);        // tile k resident; k+1..k+3 still in flight
#endif
    #pragma unroll
    for (int c = 0; c < CH; ++c) {
      float4 xx = XCHUNK(k, c);
      do_chunk(xx, st, dA, dB, dC, vout, souto, dvout, k * TT + 4 * c, decay);
    }
#if HAVE_TDM
    const int kn = k + NBUF;
    if (kn < NT)
      tdm_load(ldsOffset(&sIn[k & (NBUF - 1)][0]), gbase + (u64)kn * TT * 4u);
#endif
  }

  // ----- epilogue: dv[T-1] (no spike at t==T exists -> m <= 5) -----
  float dD = (st.m <= 5) ? 0.0f : st.px;
  float4 d4; d4.x = dA; d4.y = dB; d4.z = dC; d4.w = dD;
  *(float4*)(dvout + T_LEN - 4) = d4;
}

// ---------------------------------------------------------------------------
extern "C" void kernel_launch(void* const* d_in, const int* in_sizes, int n_in,
                              void* d_out, int out_size, void* d_ws, size_t ws_size,
                              hipStream_t stream) {
  const float* ode   = (const float*)d_in[0];   // [B, T, 1] f32
  const float* decay = (const float*)d_in[1];   // scalar f32
  float* out = (float*)d_out;                   // [v_rec | s_rec | dv], each B*T
  const int total = in_sizes[0];
  const int B = total / T_LEN;                  // 2048 for the reference shapes
  dim3 grid((unsigned)(B / ROWS)), block(ROWS);
  lif_scan_kernel<<<grid, block, 0, stream>>>(ode, decay, out, B);
}